// MaskedRoIPooling_66357244723791
// MI455X (gfx1250) — compile-verified
//
#include <hip/hip_runtime.h>

// ---------------------------------------------------------------------------
// MaskedRoIPooling on gfx1250 (MI455X), fp32 end-to-end via V_WMMA_F32_16X16X4_F32.
//
// Per (n, c): out(7,7) = Wr(7x64) * (feat[c] .* mask[n,:64,:64])(64x64) * Wc^T(64x7)
// One wave32 per (n,c); X tile staged in LDS; Wr/Wc fragments synthesized in
// registers from the box's integer bin ranges (adaptive_avg_pool2d bins).
// ---------------------------------------------------------------------------

typedef float v2f __attribute__((ext_vector_type(2)));
typedef float v8f __attribute__((ext_vector_type(8)));

#define HF 64
#define WF 64
#define OUTP 7
#define XPITCH 68   // 68*4 B = 16-byte aligned rows; bank = (4r+c)%64 -> conflict-free A reads
#define TPITCH 19   // conflict-free tmp store (D layout) and B2 reload

__global__ __launch_bounds__(32)
void roi_pool_wmma_f32(const float* __restrict__ feat,
                       const float* __restrict__ masks,
                       const float* __restrict__ boxes,
                       const int*  __restrict__ h_img_p,
                       const int*  __restrict__ w_img_p,
                       float* __restrict__ out,
                       int C)
{
    __shared__ float Xs[HF * XPITCH];   // weighted feature tile (64x64, padded)
    __shared__ float Ts[HF * TPITCH];   // tmp = X * Wc^T  (64x16, padded)

    const int blk   = blockIdx.x;
    const int n     = blk / C;
    const int c     = blk % C;
    const int lane  = threadIdx.x;
    const int idx16 = lane & 15;        // N-index (cols) / M-row within half
    const int half  = lane >> 4;        // selects K pair {0,1} vs {2,3}

    const int mh = *h_img_p;            // 1024
    const int mw = *w_img_p;            // 1024

    // ---- box -> integer bin ranges (replicates reference math) ----
    const float xsc = (float)WF / (float)mw;
    const float ysc = (float)HF / (float)mh;
    const float bx1 = boxes[n * 4 + 0], by1 = boxes[n * 4 + 1];
    const float bx2 = boxes[n * 4 + 2], by2 = boxes[n * 4 + 3];
    const float x1f = fminf(fmaxf(bx1 * xsc, 0.f), (float)(WF - 1));
    const float y1f = fminf(fmaxf(by1 * ysc, 0.f), (float)(HF - 1));
    const float x2f = fmaxf(x1f + 1.f, fminf(bx2 * xsc, (float)WF));
    const float y2f = fmaxf(y1f + 1.f, fminf(by2 * ysc, (float)HF));
    const int x1i = (int)x1f, y1i = (int)y1f;   // clamped >= 0, trunc == floor
    const int x2i = (int)x2f, y2i = (int)y2f;
    const int hh = y2i - y1i, ww = x2i - x1i;   // >= 1

    // This lane owns output bin index idx16 (valid only when < 7).
    const int bi = idx16;
    const int rs = y1i + (bi * hh) / OUTP;
    const int re = y1i + ((bi + 1) * hh + OUTP - 1) / OUTP;
    const int cs = x1i + (bi * ww) / OUTP;
    const int ce = x1i + ((bi + 1) * ww + OUTP - 1) / OUTP;
    const float rinv = (bi < OUTP) ? 1.f / (float)(re - rs) : 0.f;
    const float cinv = (bi < OUTP) ? 1.f / (float)(ce - cs) : 0.f;

    // ---- stage X = feat[c] .* mask[n,:64,:64] into LDS (b128 global loads) ----
    const float* fbase = feat  + (size_t)c * (HF * WF);
    const float* mbase = masks + (size_t)n * ((size_t)mh * (size_t)mw);
    #pragma unroll 4
    for (int it = 0; it < 32; ++it) {
        const int e = (it * 32 + lane) * 4;       // 2 rows of 64 per iteration
        const int r = e >> 6, w = e & 63;
        float4 f = *(const float4*)(fbase + e);
        float4 m = *(const float4*)(mbase + (size_t)r * mw + w);
        *(float4*)&Xs[r * XPITCH + w] =
            make_float4(f.x * m.x, f.y * m.y, f.z * m.z, f.w * m.w);
    }
    __syncthreads();

    // ---- step 1: tmp(64x16) = X(64x64) * Wc^T(64x16), 4 row-tiles x 16 K-steps ----
    v8f acc0 = {}, acc1 = {}, acc2 = {}, acc3 = {};
    #pragma unroll
    for (int kk = 0; kk < 16; ++kk) {
        const int k0 = 4 * kk + 2 * half;          // this lane's K pair
        // B fragment: Wc^T[k, j] = Wc[j=idx16, k]; synthesized, no memory.
        v2f b;
        b.x = (k0     >= cs && k0     < ce) ? cinv : 0.f;
        b.y = (k0 + 1 >= cs && k0 + 1 < ce) ? cinv : 0.f;
        // A fragments from LDS: row = 16*m0 + idx16, cols k0, k0+1.
        v2f a0 = *(const v2f*)&Xs[( 0 + idx16) * XPITCH + k0];
        v2f a1 = *(const v2f*)&Xs[(16 + idx16) * XPITCH + k0];
        v2f a2 = *(const v2f*)&Xs[(32 + idx16) * XPITCH + k0];
        v2f a3 = *(const v2f*)&Xs[(48 + idx16) * XPITCH + k0];
        acc0 = __builtin_amdgcn_wmma_f32_16x16x4_f32(false, a0, false, b, (short)0, acc0, false, false);
        acc1 = __builtin_amdgcn_wmma_f32_16x16x4_f32(false, a1, false, b, (short)0, acc1, false, false);
        acc2 = __builtin_amdgcn_wmma_f32_16x16x4_f32(false, a2, false, b, (short)0, acc2, false, false);
        acc3 = __builtin_amdgcn_wmma_f32_16x16x4_f32(false, a3, false, b, (short)0, acc3, false, false);
    }

    // ---- spill tmp to LDS in C/D layout: VGPR g -> row g + 8*half (+16*m0) ----
    __syncthreads();
    #pragma unroll
    for (int g = 0; g < 8; ++g) {
        const int rb = g + 8 * half;
        Ts[(rb +  0) * TPITCH + idx16] = acc0[g];
        Ts[(rb + 16) * TPITCH + idx16] = acc1[g];
        Ts[(rb + 32) * TPITCH + idx16] = acc2[g];
        Ts[(rb + 48) * TPITCH + idx16] = acc3[g];
    }
    __syncthreads();

    // ---- step 2: out16(16x16) = Wr_pad(16x64) * tmp(64x16), 16 K-steps ----
    v8f accO = {};
    #pragma unroll
    for (int kk = 0; kk < 16; ++kk) {
        const int k0 = 4 * kk + 2 * half;
        // A fragment: Wr[i=idx16, r = k0, k0+1]; synthesized.
        v2f a;
        a.x = (k0     >= rs && k0     < re) ? rinv : 0.f;
        a.y = (k0 + 1 >= rs && k0 + 1 < re) ? rinv : 0.f;
        // B fragment from tmp: rows k0, k0+1, col idx16.
        v2f b;
        b.x = Ts[(k0    ) * TPITCH + idx16];
        b.y = Ts[(k0 + 1) * TPITCH + idx16];
        accO = __builtin_amdgcn_wmma_f32_16x16x4_f32(false, a, false, b, (short)0, accO, false, false);
    }

    // ---- write 7x7 corner: VGPR g holds row i = g + 8*half, col j = idx16 ----
    float* obase = out + (size_t)(n * C + c) * (OUTP * OUTP);
    #pragma unroll
    for (int g = 0; g < 8; ++g) {
        const int i = g + 8 * half;
        const int j = idx16;
        if (i < OUTP && j < OUTP) obase[i * OUTP + j] = accO[g];
    }
}

extern "C" void kernel_launch(void* const* d_in, const int* in_sizes, int n_in,
                              void* d_out, int out_size, void* d_ws, size_t ws_size,
                              hipStream_t stream) {
    const float* feat  = (const float*)d_in[0];   // (1, C, 64, 64) fp32
    const float* masks = (const float*)d_in[1];   // (N, 1024, 1024) fp32
    const float* boxes = (const float*)d_in[2];   // (N, 4) fp32
    const int*   h_img = (const int*)d_in[3];     // scalar 1024
    const int*   w_img = (const int*)d_in[4];     // scalar 1024
    float*       out   = (float*)d_out;           // (N, C, 7, 7) fp32

    const int C = in_sizes[0] / (HF * WF);        // 256
    const int N = in_sizes[2] / 4;                // 64

    roi_pool_wmma_f32<<<dim3(N * C), dim3(32), 0, stream>>>(
        feat, masks, boxes, h_img, w_img, out, C);
}